// Attention_37142877176574
// MI455X (gfx1250) — compile-verified
//
#include <hip/hip_runtime.h>
#include <cmath>

// Problem constants (match reference)
#define B_   64
#define T_   1024
#define E_   512
#define DL_  1024
#define DA_  128
#define CL_  32
#define K_   31

typedef float v2f __attribute__((ext_vector_type(2)));
typedef float v8f __attribute__((ext_vector_type(8)));

// Branch-free tanh: native V_TANH_F32 on gfx1250 if the builtin exists,
// else 1 - 2/(exp(2x)+1) via fast exp (saturates correctly at +/-inf).
#if defined(__has_builtin)
#  if __has_builtin(__builtin_amdgcn_tanhf)
#    define HAVE_NATIVE_TANH 1
#  endif
#endif
static __device__ __forceinline__ float fast_tanh(float x) {
#ifdef HAVE_NATIVE_TANH
    return __builtin_amdgcn_tanhf(x);
#else
    float e = __expf(x + x);
    return 1.f - 2.f / (e + 1.f);
#endif
}

// ---------------------------------------------------------------------------
// Kernel 1: location conv  att[B,T] (*) conv_w[32,1,31], SAME pad 15.
// Bias is NOT added here (folded into the per-a bias in query_kernel).
// Output conv_out[B,T,32], c contiguous (this is the WMMA A-matrix feed).
// ---------------------------------------------------------------------------
__global__ __launch_bounds__(256)
void conv_kernel(const float* __restrict__ att,
                 const float* __restrict__ w,
                 float* __restrict__ out)
{
    __shared__ float s_in[256 + K_ - 1];
    __shared__ float s_w[CL_ * K_];
    const int b   = blockIdx.y;
    const int t0  = blockIdx.x * 256;
    const int tid = threadIdx.x;

    for (int i = tid; i < 256 + K_ - 1; i += 256) {
        int t = t0 + i - (K_ / 2);
        s_in[i] = (t >= 0 && t < T_) ? att[(size_t)b * T_ + t] : 0.f;
    }
    for (int i = tid; i < CL_ * K_; i += 256) s_w[i] = w[i];
    __syncthreads();

    float* orow = out + ((size_t)b * T_ + t0 + tid) * CL_;
    #pragma unroll 4
    for (int c = 0; c < CL_; ++c) {
        float s = 0.f;
        #pragma unroll
        for (int k = 0; k < K_; ++k) s = fmaf(s_in[tid + k], s_w[c * K_ + k], s);
        orow[c] = s;
    }
}

// ---------------------------------------------------------------------------
// Kernel 2: combined query projection + folded biases.
// qc[b,a] = sum_d lstm[b,d]*W_lstm[a,d] + b_loc[a] + sum_c conv_b[c]*W_loc[a,c]
// ---------------------------------------------------------------------------
__global__ __launch_bounds__(DA_)
void query_kernel(const float* __restrict__ lstm,
                  const float* __restrict__ W_lstm,
                  const float* __restrict__ W_loc,
                  const float* __restrict__ conv_b,
                  const float* __restrict__ b_loc,
                  float* __restrict__ qc)
{
    __shared__ float s_lstm[DL_];
    const int b = blockIdx.x;
    const int a = threadIdx.x;                 // 128 threads
    for (int i = a; i < DL_; i += DA_) s_lstm[i] = lstm[(size_t)b * DL_ + i];
    __syncthreads();

    float s = b_loc[a];
    #pragma unroll
    for (int c = 0; c < CL_; ++c) s = fmaf(conv_b[c], W_loc[a * CL_ + c], s);
    const float* wr = W_lstm + (size_t)a * DL_;
    for (int d = 0; d < DL_; ++d) s = fmaf(s_lstm[d], wr[d], s);
    qc[(size_t)b * DA_ + a] = s;
}

// ---------------------------------------------------------------------------
// Kernel 3: fused  loc-GEMM (WMMA f32 16x16x4) + tanh + W_e dot.
// One wave owns a 16-row t-tile: acc[t0:t0+16, 0:128] = conv_tile @ W_loc^T,
// then e[t] = sum_a tanh(acc + qc[b,a] + pe[b,t,a]) * W_e[a] + b_e.
// A layout (16x4 f32): lane%16 = M, VGPR0 = K(2*(lane/16)), VGPR1 = K+1.
// B layout (4x16 f32): lane%16 = N, VGPR0 = K(2*(lane/16)), VGPR1 = K+1.
// C/D layout: N = lane%16, M = vgpr + 8*(lane/16).
// ---------------------------------------------------------------------------
__global__ __launch_bounds__(128)
void energies_kernel(const float* __restrict__ conv_out,
                     const float* __restrict__ W_loc,
                     const float* __restrict__ qc,
                     const float* __restrict__ pe,
                     const float* __restrict__ W_e,
                     const float* __restrict__ b_e,
                     float* __restrict__ energies)
{
    __shared__ float sW[DA_ * CL_];   // W_loc [a][c], 16 KB
    __shared__ float sWe[DA_];
    const int b   = blockIdx.y;
    const int tid = threadIdx.x;      // 128 threads = 4 waves
    for (int i = tid; i < DA_ * CL_; i += 128) sW[i] = W_loc[i];
    if (tid < DA_) sWe[tid] = W_e[tid];
    __syncthreads();

    const int wave = tid >> 5;
    const int lane = tid & 31;
    const int half = lane >> 4;       // 0: lanes 0-15, 1: lanes 16-31
    const int l    = lane & 15;
    const int t0   = (blockIdx.x * 4 + wave) * 16;

    // A fragments: conv_out tile [16 t x 32 c], 8 K-steps of 4.
    v2f afrag[8];
    const float* ar = conv_out + ((size_t)b * T_ + t0 + l) * CL_ + half * 2;
    #pragma unroll
    for (int kk = 0; kk < 8; ++kk) {
        afrag[kk].x = ar[kk * 4];
        afrag[kk].y = ar[kk * 4 + 1];
    }

    const float be = b_e[0];
    float psum[8];
    #pragma unroll
    for (int r = 0; r < 8; ++r) psum[r] = 0.f;

    const float* peb = pe + ((size_t)b * T_ + t0) * DA_;

    #pragma unroll
    for (int n = 0; n < 8; ++n) {
        const int a = n * 16 + l;
        v8f acc = {0.f, 0.f, 0.f, 0.f, 0.f, 0.f, 0.f, 0.f};
        const float* br = sW + a * CL_ + half * 2;   // B[c][a] = W_loc[a][c]
        #pragma unroll
        for (int kk = 0; kk < 8; ++kk) {
            v2f bfrag;
            bfrag.x = br[kk * 4];
            bfrag.y = br[kk * 4 + 1];
            acc = __builtin_amdgcn_wmma_f32_16x16x4_f32(
                false, afrag[kk], false, bfrag, (short)0, acc, false, false);
        }
        const float wq = qc[(size_t)b * DA_ + a];
        const float we = sWe[a];
        #pragma unroll
        for (int r = 0; r < 8; ++r) {
            const int m = r + half * 8;              // t row within tile
            float v = fast_tanh(acc[r] + wq + peb[(size_t)m * DA_ + a]);
            psum[r] = fmaf(v, we, psum[r]);
        }
    }

    // Reduce over a (16 lanes of each half hold N=0..15 for a fixed M row).
    #pragma unroll
    for (int r = 0; r < 8; ++r) {
        float s = psum[r];
        s += __shfl_xor(s, 1, 32);
        s += __shfl_xor(s, 2, 32);
        s += __shfl_xor(s, 4, 32);
        s += __shfl_xor(s, 8, 32);                   // stays within 16-lane half
        if (l == 0)
            energies[(size_t)b * T_ + t0 + half * 8 + r] = s + be;
    }
}

// ---------------------------------------------------------------------------
// Kernel 4: per-row softmax over T, writes weights directly to output slot.
// ---------------------------------------------------------------------------
__global__ __launch_bounds__(256)
void softmax_kernel(const float* __restrict__ energies,
                    float* __restrict__ weights)
{
    __shared__ float red[256];
    const int b   = blockIdx.x;
    const int tid = threadIdx.x;
    const float* e = energies + (size_t)b * T_;
    float m = -INFINITY;
    for (int t = tid; t < T_; t += 256) m = fmaxf(m, e[t]);
    red[tid] = m; __syncthreads();
    for (int s = 128; s > 0; s >>= 1) {
        if (tid < s) red[tid] = fmaxf(red[tid], red[tid + s]);
        __syncthreads();
    }
    const float mx = red[0];
    __syncthreads();

    float acc = 0.f;
    float* w = weights + (size_t)b * T_;
    for (int t = tid; t < T_; t += 256) {
        float ex = __expf(e[t] - mx);
        w[t] = ex;
        acc += ex;
    }
    red[tid] = acc; __syncthreads();
    for (int s = 128; s > 0; s >>= 1) {
        if (tid < s) red[tid] += red[tid + s];
        __syncthreads();
    }
    const float inv = 1.f / red[0];
    for (int t = tid; t < T_; t += 256) w[t] *= inv;
}

// ---------------------------------------------------------------------------
// Kernel 5: context partials. Streams encoder_output (134 MB) once,
// coalesced over 64 consecutive e per quarter-block, t split 4x4.
// part[b,e,ts] = sum over 256 t of w[b,t]*enc[b,t,e]
// ---------------------------------------------------------------------------
__global__ __launch_bounds__(256)
void context_partial_kernel(const float* __restrict__ weights,
                            const float* __restrict__ enc,
                            float* __restrict__ part)
{
    __shared__ float sw[256];
    __shared__ float sp[256];
    const int b      = blockIdx.z;
    const int e      = blockIdx.x * 64 + (threadIdx.x & 63);
    const int tg     = threadIdx.x >> 6;          // 0..3
    const int tchunk = blockIdx.y * 256;

    sw[threadIdx.x] = weights[(size_t)b * T_ + tchunk + threadIdx.x];
    __syncthreads();

    const float* er = enc + ((size_t)b * T_ + tchunk + tg * 64) * E_ + e;
    float acc = 0.f;
    #pragma unroll 4
    for (int i = 0; i < 64; ++i)
        acc = fmaf(sw[tg * 64 + i], er[(size_t)i * E_], acc);

    sp[threadIdx.x] = acc;
    __syncthreads();
    if (tg == 0) {
        float s = sp[threadIdx.x] + sp[threadIdx.x + 64] +
                  sp[threadIdx.x + 128] + sp[threadIdx.x + 192];
        part[(((size_t)b * E_) + e) * 4 + blockIdx.y] = s;
    }
}

__global__ __launch_bounds__(256)
void context_reduce_kernel(const float* __restrict__ part,
                           float* __restrict__ ctx)
{
    const int i = blockIdx.x * 256 + threadIdx.x;  // < B*E
    const float* p = part + (size_t)i * 4;
    ctx[i] = p[0] + p[1] + p[2] + p[3];
}

// ---------------------------------------------------------------------------
extern "C" void kernel_launch(void* const* d_in, const int* in_sizes, int n_in,
                              void* d_out, int out_size, void* d_ws, size_t ws_size,
                              hipStream_t stream)
{
    (void)in_sizes; (void)n_in; (void)out_size; (void)ws_size;

    const float* enc  = (const float*)d_in[0];   // [B,T,E]
    const float* pe   = (const float*)d_in[1];   // [B,T,128]
    const float* lstm = (const float*)d_in[2];   // [B,1,1024]
    const float* awc  = (const float*)d_in[3];   // [B,T]
    const float* Wl   = (const float*)d_in[4];   // [128,1024]
    const float* cw   = (const float*)d_in[5];   // [32,1,31]
    const float* cb   = (const float*)d_in[6];   // [32]
    const float* Wloc = (const float*)d_in[7];   // [128,32]
    const float* bloc = (const float*)d_in[8];   // [128]
    const float* We   = (const float*)d_in[9];   // [1,128]
    const float* be   = (const float*)d_in[10];  // [1]

    float* out_ctx = (float*)d_out;                       // [B,1,E]
    float* out_w   = out_ctx + (size_t)B_ * E_;           // [B,T]

    float* ws       = (float*)d_ws;
    float* conv_out = ws;                                  // B*T*32
    float* qc       = conv_out + (size_t)B_ * T_ * CL_;    // B*128
    float* energies = qc + (size_t)B_ * DA_;               // B*T
    float* part     = energies + (size_t)B_ * T_;          // B*E*4

    conv_kernel<<<dim3(T_ / 256, B_), 256, 0, stream>>>(awc, cw, conv_out);
    query_kernel<<<B_, DA_, 0, stream>>>(lstm, Wl, Wloc, cb, bloc, qc);
    energies_kernel<<<dim3(T_ / 64, B_), 128, 0, stream>>>(conv_out, Wloc, qc,
                                                           pe, We, be, energies);
    softmax_kernel<<<B_, 256, 0, stream>>>(energies, out_w);
    context_partial_kernel<<<dim3(E_ / 64, 4, B_), 256, 0, stream>>>(out_w, enc, part);
    context_reduce_kernel<<<(B_ * E_) / 256, 256, 0, stream>>>(part, out_ctx);
}